// DiffusionLayer_50491635532440
// MI455X (gfx1250) — compile-verified
//
#include <hip/hip_runtime.h>
#include <stdint.h>

// ---------------------------------------------------------------------------
// ADI diffusion (Thomas tridiagonal solves), fully fused in one kernel.
// One workgroup per (batch, channel) slice; u tile + coeff/cs workspace in LDS
// with a 129-float row stride (bank-conflict-free along BOTH solve directions
// on 64 banks, wave32). Tile is loaded by the CDNA5 Tensor Data Mover with
// pad_enable producing the 129-stride layout directly.
// ---------------------------------------------------------------------------

namespace {
constexpr int   kS     = 128;
constexpr int   kB     = 32;
constexpr int   kC     = 3;
constexpr int   kSteps = 10;
constexpr float kDT    = 0.001f;
constexpr float kEPS   = 1e-6f;
constexpr int   kLS    = kS + 1;   // LDS row stride in floats (129)
}

typedef unsigned int u32;
typedef unsigned int v4u __attribute__((ext_vector_type(4)));
typedef int          v4i __attribute__((ext_vector_type(4)));
typedef int          v8i __attribute__((ext_vector_type(8)));

#if defined(__has_builtin)
#if __has_builtin(__builtin_amdgcn_tensor_load_to_lds)
#define HAVE_TDM 1
#endif
#if __has_builtin(__builtin_amdgcn_s_wait_tensorcnt)
#define HAVE_WAIT_TENSOR 1
#endif
#endif

// One complete phase for this thread's line L:
//   dir==0: solve along w on row L   (coeff smoothed along w)
//   dir==1: solve along h on col L   (coeff smoothed along h)
// W row L is thread-private in every phase; only U crosses threads, so a
// single barrier (before touching U) per phase suffices.
__device__ __forceinline__ void thomas_phase(
    float* __restrict__ Ush, float* __restrict__ Wsh,
    const float* __restrict__ cbase, const float* __restrict__ ctc,
    int L, float t, float dtp, int dir)
{
  // 1) coefficient line: W[L][i] = (dtp/3)*(v[i-1]+v[i]+v[i+1]),
  //    v = max(base + tc*t, EPS), edge-replicated (matches reference smooth).
  const float scale = dtp * (1.0f / 3.0f);
  const int g0  = (dir == 0) ? L * kS : L;
  const int gst = (dir == 0) ? 1 : kS;
  float* Wl = Wsh + L * kLS;
  float vm, vc;
  {
    const float v0 = fmaxf(cbase[g0] + ctc[g0] * t, kEPS);
    vm = v0; vc = v0;
  }
#pragma unroll 4
  for (int i = 0; i < kS; ++i) {
    float vp;
    if (i < kS - 1) {
      const int g = g0 + (i + 1) * gst;
      vp = fmaxf(cbase[g] + ctc[g] * t, kEPS);
    } else {
      vp = vc;                       // replicate edge
    }
    Wl[i] = scale * (vm + vc + vp);
    vm = vc; vc = vp;
  }

  // previous phase wrote U in the other orientation -> barrier before use
  __syncthreads();

  float*   Up = (dir == 0) ? (Ush + L * kLS) : (Ush + L);
  const int us = (dir == 0) ? 1 : kLS;

  // 2) forward sweep: overwrite W with cs, U with ds (in place)
  float csp, dsp;
  {
    const float k0  = Wl[0];
    const float den = (1.0f + k0) + kEPS;     // b0 = 1 + coeff0 (endpoint)
    csp = (-k0) / den;                        // c0 / den
    dsp = Up[0] / den;
    Wl[0] = csp;
    Up[0] = dsp;
  }
#pragma unroll 2
  for (int i = 1; i < kS; ++i) {
    const float ki = Wl[i];
    const float ai = -ki;
    const float bi = (i == kS - 1) ? (1.0f + ki) : (1.0f + 2.0f * ki);
    const float den = (bi - ai * csp) + kEPS;
    const float cs = (-ki) / den;                     // two true divisions,
    const float ds = (Up[i * us] - ai * dsp) / den;   // matching reference
    Wl[i] = cs;
    Up[i * us] = ds;
    csp = cs; dsp = ds;
  }

  // 3) back substitution in place
  float x = dsp;                                // x[N-1] = ds[N-1]
#pragma unroll 4
  for (int i = kS - 2; i >= 0; --i) {
    x = Up[i * us] - Wl[i] * x;
    Up[i * us] = x;
  }
  // next phase's barrier orders these U writes before cross-thread reads
}

__global__ void __launch_bounds__(kS, 1)
adi_diffusion_kernel(const float* __restrict__ u_in,
                     const float* __restrict__ a_base,
                     const float* __restrict__ b_base,
                     const float* __restrict__ a_tc,
                     const float* __restrict__ b_tc,
                     float* __restrict__ out)
{
  __shared__ float Ush[kS * kLS];   // state tile, 129-float row stride
  __shared__ float Wsh[kS * kLS];   // coeff -> cs workspace

  const int tid = threadIdx.x;      // line index L (one line per thread)
  const int bc  = blockIdx.x;       // b*C + c
  const int c   = bc % kC;
  const float* gu = u_in + (size_t)bc * kS * kS;

#if defined(HAVE_TDM)
  // --- Tensor Data Mover: DMA the whole 128x128 tile into LDS with
  //     pad_enable = 1 DWORD pad every 128 DWORDs -> 129-float rows. ---
  if (tid < 32) {                   // single wave issues the TDM op
    const u32 lds_off = (u32)(size_t)(__attribute__((address_space(3))) float*)&Ush[0];
    const uint64_t ga = (uint64_t)(uintptr_t)gu;
    v4u g0;
    g0.x = 1u;                                            // count=1, user desc
    g0.y = lds_off;                                       // lds_addr (bytes)
    g0.z = (u32)ga;                                       // global_addr[31:0]
    g0.w = (u32)((ga >> 32) & 0x01FFFFFFu) | (2u << 30);  // addr[56:32] | type=2
    v8i g1;
    g1[0] = (int)((2u << 16)      // data_size = 4 bytes
                | (1u << 20)      // pad_enable
                | (6u << 22));    // pad_interval: 128 DWORDs; pad_amount=0 -> 1 DWORD
    g1[1] = (int)(128u << 16);    // tensor_dim0 = 128
    g1[2] = (int)(128u << 16);    // tensor_dim1 = 128
    g1[3] = (int)(128u << 16);    // tile_dim0   = 128
    g1[4] = 128;                  // tile_dim1 = 128, tile_dim2 = 0
    g1[5] = 128;                  // tensor_dim0_stride = 128
    g1[6] = 0;
    g1[7] = 0;
    v4i gz = {0, 0, 0, 0};
#if (__clang_major__ >= 23)
    v8i gz8 = {0, 0, 0, 0, 0, 0, 0, 0};
    __builtin_amdgcn_tensor_load_to_lds(g0, g1, gz, gz, gz8, 0);
#else
    __builtin_amdgcn_tensor_load_to_lds(g0, g1, gz, gz, 0);
#endif
#if defined(HAVE_WAIT_TENSOR)
    __builtin_amdgcn_s_wait_tensorcnt(0);
#else
    asm volatile("s_wait_tensorcnt 0" ::: "memory");
#endif
  }
  __syncthreads();
#else
  // Fallback: coalesced plain loads into the padded layout.
  for (int j = tid; j < kS * kS; j += kS) {
    const int h = j >> 7, w = j & (kS - 1);
    Ush[h * kLS + w] = gu[j];
  }
  __syncthreads();
#endif

  const size_t coff = (size_t)c * kS * kS;
  const float* ab = a_base + coff;
  const float* at = a_tc   + coff;
  const float* bb = b_base + coff;
  const float* bt = b_tc   + coff;

#pragma unroll 1
  for (int n = 0; n < kSteps; ++n) {
    const float t0 = (float)n * kDT;
    thomas_phase(Ush, Wsh, ab, at, tid, t0,               0.5f * kDT, 0); // x, dt/2
    thomas_phase(Ush, Wsh, bb, bt, tid, t0 + 0.5f * kDT,  kDT,        1); // y, dt
    thomas_phase(Ush, Wsh, ab, at, tid, t0 + kDT,         0.5f * kDT, 0); // x, dt/2
  }

  __syncthreads();   // last x-phase wrote rows; write back columns coalesced
  float* gout = out + (size_t)bc * kS * kS;
#pragma unroll 4
  for (int h = 0; h < kS; ++h) {
    gout[h * kS + tid] = Ush[h * kLS + tid];
  }
}

extern "C" void kernel_launch(void* const* d_in, const int* in_sizes, int n_in,
                              void* d_out, int out_size, void* d_ws, size_t ws_size,
                              hipStream_t stream) {
  (void)in_sizes; (void)n_in; (void)d_ws; (void)ws_size; (void)out_size;
  const float* u  = (const float*)d_in[0];
  const float* ab = (const float*)d_in[1];
  const float* bb = (const float*)d_in[2];
  const float* at = (const float*)d_in[3];
  const float* bt = (const float*)d_in[4];
  dim3 grid(kB * kC);
  dim3 block(kS);
  adi_diffusion_kernel<<<grid, block, 0, stream>>>(u, ab, bb, at, bt, (float*)d_out);
}